// SoftAUAvULoss_78769700209211
// MI455X (gfx1250) — compile-verified
//
#include <hip/hip_runtime.h>
#include <stdint.h>

// ---------------- problem constants (match reference) ----------------
#define NSAMP 4194304
#define CCH 10
#define ROWS_PER_TILE 256
#define ELEMS_PER_TILE (ROWS_PER_TILE * CCH)   // 2560 f32 = 10240 B
#define NTILES (NSAMP / ROWS_PER_TILE)         // 16384
#define NBLK_A 2048
#define TILES_PER_BLK (NTILES / NBLK_A)        // 8
#define NBLK_B 2048
#define BATCH_PER_BLK (NTILES / NBLK_B)        // 8
#define NBUCKET 22                             // buckets 0..20 plus "above all thresholds"
#define HISTF (NBUCKET * 4)                    // 88 floats per B-block partial
#define NBUF 4                                 // TDM pipeline depth (tiles in flight)
#define WAVE_ELEMS (32 * CCH)                  // 320 f32 per wave slice = 1280 B

#define ENTMAX 2.30258509299404568402f         // ln(10), rounds to f32(ln 10)
#define LOG_1EM8 -18.420680743952367f          // ln(1e-8)

typedef __attribute__((ext_vector_type(2))) float v2f;
typedef __attribute__((ext_vector_type(8))) float v8f;
typedef __attribute__((ext_vector_type(4))) unsigned int v4u;
typedef __attribute__((ext_vector_type(4))) int v4i;
typedef __attribute__((ext_vector_type(8))) int v8i;

// ---------------- ws layout (bytes); total ~21.8 MB ----------------
#define WS_UNC   0ull                                   // NSAMP f32
#define WS_ACC   (WS_UNC + (size_t)NSAMP * 4)           // NSAMP u8
#define WS_BPART (WS_ACC + (size_t)NSAMP)               // NBLK_A * 4 f32
#define WS_SCAL  (WS_BPART + (size_t)NBLK_A * 16)       // 16 f32 (focal,ce,umin,umax,label0)
#define WS_HIST  (WS_SCAL + 64)                         // NBLK_B * 88 f32

// LDS byte offset of a __shared__ object: generic LDS pointers are
// {aperture_hi32 | offset_lo32} (ISA 10.2), so the low 32 bits are the
// wave-relative LDS byte address the TDM D# wants.
__device__ inline unsigned int lds_off(const void* p) {
  return (unsigned int)(uintptr_t)p;
}

// TDM 1-D tile load: contiguous nelem f32 from global -> LDS (ISA ch.8 D#).
__device__ inline void tdm_load_1d(const void* gsrc, unsigned int ldsByteOff,
                                   unsigned int nelem) {
  unsigned long long ga = (unsigned long long)(uintptr_t)gsrc;
  v4u g0;
  g0.x = 1u;                                   // count=1 (valid), user mode
  g0.y = ldsByteOff;                           // lds_addr (bytes)
  g0.z = (unsigned int)ga;                     // global_addr[31:0]
  g0.w = (unsigned int)((ga >> 32) & 0x1FFFFFFull) | (2u << 30); // addr[56:32] | type=2
  v8i g1;
  g1[0] = (int)(2u << 16);                     // wg_mask=0 (not in cluster), data_size=2 (4B)
  g1[1] = (int)(nelem << 16);                  // atomic_barrier=0 | tensor_dim0[15:0]
  g1[2] = (int)((nelem >> 16) | (1u << 16));   // tensor_dim0[31:16] | tensor_dim1[15:0]=1
  g1[3] = (int)(nelem << 16);                  // tensor_dim1[31:16]=0 | tile_dim0
  g1[4] = 1;                                   // tile_dim1=1 | tile_dim2=0
  g1[5] = (int)nelem;                          // tensor_dim0_stride[31:0]
  g1[6] = 0;                                   // stride0[47:32] | stride1[15:0]
  g1[7] = 0;                                   // stride1[47:16]
  v4i gz = {0, 0, 0, 0};
#if defined(__clang_major__) && (__clang_major__ >= 23)
  v8i gz8 = {0, 0, 0, 0, 0, 0, 0, 0};
  __builtin_amdgcn_tensor_load_to_lds(g0, g1, gz, gz, gz8, 0);
#else
  __builtin_amdgcn_tensor_load_to_lds(g0, g1, gz, gz, 0);
#endif
}

// ---------------- kernel 0: label0 = flat argmax(y) == argmax of row 0 ----
__global__ void kInit(const float* __restrict__ y, float* __restrict__ scal) {
  if (threadIdx.x == 0) {
    float best = y[0];
    int idx = 0;
    for (int j = 1; j < CCH; ++j) {
      float v = y[j];
      if (v > best) { best = v; idx = j; }  // first-max kept, like jnp.argmax
    }
    scal[4] = (float)idx;
  }
}

// ---------------- kernel A: TDM-staged streaming pass over probs & y ------
// Each wave TDM-loads its OWN 32-row slice of every tile (1280 B per array),
// 4 tiles deep, so no workgroup barriers are needed in the hot loop and the
// TDM issue runs with full EXEC in every wave (TDM ignores EXEC, so it must
// not sit in a lane-masked region).
// Per sample: unc (entropy), acc (argmax==label0), focal/ce partial sums,
// per-thread min/max of unc. Block partials -> ws (no atomics).
__global__ __launch_bounds__(256) void kA(const float* __restrict__ probs,
                                          const float* __restrict__ y,
                                          float* __restrict__ unc,
                                          unsigned char* __restrict__ accb,
                                          float* __restrict__ bpart,
                                          const float* __restrict__ scal) {
  __shared__ __align__(16) float sP[NBUF][ELEMS_PER_TILE];
  __shared__ __align__(16) float sY[NBUF][ELEMS_PER_TILE];
  __shared__ float sRed[8 * 4];

  const int tid = threadIdx.x;
  const int wave = tid >> 5;                 // wave-uniform
  const int lane = tid & 31;
  const int label0 = (int)scal[4];

  float focal = 0.f, ce = 0.f;
  float umin = 3.402823466e38f, umax = -3.402823466e38f;

  const int tile0 = blockIdx.x * TILES_PER_BLK;
  const unsigned int sliceOff = (unsigned int)wave * WAVE_ELEMS;  // elems into tile

  // prologue: prefetch tiles 0..NBUF-2 for this wave's slice
#pragma unroll
  for (int pt = 0; pt < NBUF - 1; ++pt) {
    const size_t g = (size_t)(tile0 + pt) * ELEMS_PER_TILE + sliceOff;
    tdm_load_1d(probs + g, lds_off(&sP[pt][sliceOff]), WAVE_ELEMS);
    tdm_load_1d(y + g,     lds_off(&sY[pt][sliceOff]), WAVE_ELEMS);
  }

  for (int t = 0; t < TILES_PER_BLK; ++t) {
    const int buf = t & (NBUF - 1);
    if (t + NBUF - 1 < TILES_PER_BLK) {      // keep pipeline NBUF-1 tiles deep
      const size_t g = (size_t)(tile0 + t + NBUF - 1) * ELEMS_PER_TILE + sliceOff;
      tdm_load_1d(probs + g, lds_off(&sP[(t + NBUF - 1) & (NBUF - 1)][sliceOff]), WAVE_ELEMS);
      tdm_load_1d(y + g,     lds_off(&sY[(t + NBUF - 1) & (NBUF - 1)][sliceOff]), WAVE_ELEMS);
      __builtin_amdgcn_s_wait_tensorcnt(6);  // in-order: oldest 2 (tile t) done
    } else if (t + NBUF - 2 < TILES_PER_BLK) {
      __builtin_amdgcn_s_wait_tensorcnt(4);
    } else if (t + NBUF - 3 < TILES_PER_BLK) {
      __builtin_amdgcn_s_wait_tensorcnt(2);
    } else {
      __builtin_amdgcn_s_wait_tensorcnt(0);
    }

    const float* pr = &sP[buf][tid * CCH];   // tid*CCH lies inside this wave's slice
    const float* yr = &sY[buf][tid * CCH];
    float u = 0.f, cs = 0.f, fs = 0.f;
    int pred = 0;
    float best = pr[0];
#pragma unroll
    for (int j = 0; j < CCH; ++j) {
      float p = pr[j];
      if (j > 0 && p > best) { best = p; pred = j; }
      float l10 = logf(fmaxf(p, 1e-10f));           // log(max(p,1e-10))
      float l8 = (p >= 1e-8f) ? l10 : LOG_1EM8;     // == log(max(p,1e-8))
      u -= p * l10;
      float yv = yr[j];
      cs += yv * l8;                 // ce_per = -cs
      fs += yv * l8 * (1.0f - p);    // focal_per = -fs
    }
    const int row = (tile0 + t) * ROWS_PER_TILE + tid;
    unc[row] = u;
    accb[row] = (pred == label0) ? 1u : 0u;
    umin = fminf(umin, u);
    umax = fmaxf(umax, u);
    focal += fs;
    ce += cs;
    // all LDS reads of this buffer retired before the TDM that refills it
    asm volatile("s_wait_dscnt 0x0" ::: "memory");
  }

  // Wave reduction of focal/ce with V_WMMA_F32_16X16X4_F32, all-ones B.
  // A layout (ISA 7.12.2): lanes 0-15 supply A[m=lane, k=0/1] in vgpr0/1,
  // lanes 16-31 supply A[m=lane-16, k=2/3]. With B all-ones:
  // D[m,n] = partial[m] + partial[m+16] for every n (exact f32 MAC).
  v2f af; af.x = focal; af.y = 0.0f;
  v2f ac; ac.x = ce;    ac.y = 0.0f;
  v2f ones; ones.x = 1.0f; ones.y = 1.0f;
  v8f df = {0.f, 0.f, 0.f, 0.f, 0.f, 0.f, 0.f, 0.f};
  v8f dc = {0.f, 0.f, 0.f, 0.f, 0.f, 0.f, 0.f, 0.f};
  df = __builtin_amdgcn_wmma_f32_16x16x4_f32(false, af, false, ones, (short)0, df, false, false);
  dc = __builtin_amdgcn_wmma_f32_16x16x4_f32(false, ac, false, ones, (short)0, dc, false, false);
  // lanes 0-15 hold rows 0..7 in components 0..7; lanes 16-31 hold rows 8..15
  float sf = df[0] + df[1] + df[2] + df[3] + df[4] + df[5] + df[6] + df[7];
  float sc = dc[0] + dc[1] + dc[2] + dc[3] + dc[4] + dc[5] + dc[6] + dc[7];
  sf += __shfl_xor(sf, 16);
  sc += __shfl_xor(sc, 16);
#pragma unroll
  for (int o = 16; o > 0; o >>= 1) {
    umin = fminf(umin, __shfl_xor(umin, o));
    umax = fmaxf(umax, __shfl_xor(umax, o));
  }
  if (lane == 0) {
    sRed[wave * 4 + 0] = sf;
    sRed[wave * 4 + 1] = sc;
    sRed[wave * 4 + 2] = umin;
    sRed[wave * 4 + 3] = umax;
  }
  __syncthreads();
  if (tid == 0) {  // fixed-order cross-wave combine -> deterministic
    float F = 0.f, Cc = 0.f, mn = sRed[2], mx = sRed[3];
    for (int w = 0; w < 8; ++w) {
      F += sRed[w * 4 + 0];
      Cc += sRed[w * 4 + 1];
      mn = fminf(mn, sRed[w * 4 + 2]);
      mx = fmaxf(mx, sRed[w * 4 + 3]);
    }
    bpart[blockIdx.x * 4 + 0] = F;
    bpart[blockIdx.x * 4 + 1] = Cc;
    bpart[blockIdx.x * 4 + 2] = mn;
    bpart[blockIdx.x * 4 + 3] = mx;
  }
}

// ---------------- kernel R1: deterministic reduce of A-block partials -----
__global__ __launch_bounds__(256) void kR1(const float* __restrict__ bpart,
                                           float* __restrict__ scal) {
  __shared__ float sF[256], sC[256], sMn[256], sMx[256];
  const int tid = threadIdx.x;
  float f = 0.f, c = 0.f, mn = 3.402823466e38f, mx = -3.402823466e38f;
  for (int i = tid; i < NBLK_A; i += 256) {  // fixed ascending order
    f += bpart[i * 4 + 0];
    c += bpart[i * 4 + 1];
    mn = fminf(mn, bpart[i * 4 + 2]);
    mx = fmaxf(mx, bpart[i * 4 + 3]);
  }
  sF[tid] = f; sC[tid] = c; sMn[tid] = mn; sMx[tid] = mx;
  __syncthreads();
  for (int s = 128; s > 0; s >>= 1) {
    if (tid < s) {
      sF[tid] += sF[tid + s];
      sC[tid] += sC[tid + s];
      sMn[tid] = fminf(sMn[tid], sMn[tid + s]);
      sMx[tid] = fmaxf(sMx[tid], sMx[tid + s]);
    }
    __syncthreads();
  }
  if (tid == 0) {
    scal[0] = sF[0];
    scal[1] = sC[0];
    scal[2] = sMn[0];
    scal[3] = sMx[0];
  }
}

// ---------------- kernel B: bucketed (22-bin) weighted histogram ----------
// bucket(i) = min{k : unc[i] <= umin + th_k*(umax-umin)}, 21 if none.
// n_ac[k] etc. become prefix/suffix sums of these bins (exact restructuring).
__global__ __launch_bounds__(256) void kB(const float* __restrict__ unc,
                                          const unsigned char* __restrict__ accb,
                                          const float* __restrict__ scal,
                                          float* __restrict__ hist) {
  __shared__ int sBkt[256];
  __shared__ float sVal[256][4];
  const int tid = threadIdx.x;
  const float umin = scal[2];
  const float umax = scal[3];
  const float range = umax - umin;

  float h0 = 0.f, h1 = 0.f, h2 = 0.f, h3 = 0.f;  // owner-thread bin sums

  for (int q = 0; q < BATCH_PER_BLK; ++q) {
    const int i = (blockIdx.x * BATCH_PER_BLK + q) * 256 + tid;
    const float u = unc[i];
    const float a = accb[i] ? 1.0f : 0.0f;

    const float e = u / ENTMAX;
    const float num = e * (1.0f - 0.1f);
    const float den = fmaxf((1.0f - e) * 0.1f, 1e-10f);
    const float frac = fmaxf(num / den, 1e-10f);
    const float x = logf(frac) * 100.0f;           // log(frac)/TEMP
    const float s = 1.0f / (1.0f + expf(-x));      // sigmoid (sat to 0/1 like jax)
    const float t = tanhf(u);
    const float wc = (1.0f - s) * (1.0f - t);
    const float wu = s * t;

    int b = 21;
    for (int k = 0; k < 21; ++k) {                 // same fp expr as reference
      const float th = umin + ((float)k * 0.05f) * range;
      if (u <= th) { b = k; break; }
    }
    sBkt[tid] = b;
    sVal[tid][0] = a * wc;
    sVal[tid][1] = a * wu;
    sVal[tid][2] = (1.0f - a) * wc;
    sVal[tid][3] = (1.0f - a) * wu;
    __syncthreads();
    if (tid < NBUCKET) {                           // fixed-order scan -> deterministic
      for (int j = 0; j < 256; ++j) {
        if (sBkt[j] == tid) {
          h0 += sVal[j][0];
          h1 += sVal[j][1];
          h2 += sVal[j][2];
          h3 += sVal[j][3];
        }
      }
    }
    __syncthreads();
  }
  if (tid < NBUCKET) {
    float* o = hist + (size_t)blockIdx.x * HISTF + tid * 4;
    o[0] = h0; o[1] = h1; o[2] = h2; o[3] = h3;
  }
}

// ---------------- kernel C: finalize (prefix sums, AvU, AUC, losses) ------
__global__ __launch_bounds__(128) void kC(const float* __restrict__ hist,
                                          const float* __restrict__ scal,
                                          float* __restrict__ out) {
  __shared__ float h[HISTF];
  const int tid = threadIdx.x;
  if (tid < HISTF) {
    float s = 0.f;
    for (int b = 0; b < NBLK_B; ++b) s += hist[(size_t)b * HISTF + tid];  // fixed order
    h[tid] = s;
  }
  __syncthreads();
  if (tid == 0) {
    float T_au = 0.f, T_iu = 0.f;
    for (int b = 0; b < NBUCKET; ++b) { T_au += h[b * 4 + 1]; T_iu += h[b * 4 + 3]; }
    float p_ac = 0.f, p_au = 0.f, p_ic = 0.f, p_iu = 0.f;
    float avu[21];
    for (int k = 0; k < 21; ++k) {
      p_ac += h[k * 4 + 0];
      p_au += h[k * 4 + 1];
      p_ic += h[k * 4 + 2];
      p_iu += h[k * 4 + 3];
      const float n_ac = p_ac, n_ic = p_ic;
      const float n_au = T_au - p_au;              // gt = 1 - le
      const float n_iu = T_iu - p_iu;
      avu[k] = (n_ac + n_iu) / (n_ac + n_au + n_ic + n_iu + 1e-10f);
    }
    float auc = 0.f;
    for (int k = 0; k < 20; ++k) {
      const float dx = (float)(k + 1) * 0.05f - (float)k * 0.05f;
      auc += (avu[k + 1] + avu[k]) * 0.5f * dx;
    }
    const float focal_loss = -scal[0] / (float)NSAMP;
    const float ce_loss = -scal[1] / (float)NSAMP;
    const float savu = -1.0f * logf(fmaxf(auc, 1e-10f)) + focal_loss;
    out[0] = savu;
    out[1] = ce_loss;
  }
}

// ---------------- launch --------------------------------------------------
extern "C" void kernel_launch(void* const* d_in, const int* in_sizes, int n_in,
                              void* d_out, int out_size, void* d_ws, size_t ws_size,
                              hipStream_t stream) {
  (void)in_sizes; (void)n_in; (void)out_size; (void)ws_size;
  const float* probs = (const float*)d_in[0];
  const float* y = (const float*)d_in[1];
  float* out = (float*)d_out;
  char* ws = (char*)d_ws;  // needs ~21.8 MB; every consumed slot is rewritten each call
  float* unc = (float*)(ws + WS_UNC);
  unsigned char* accb = (unsigned char*)(ws + WS_ACC);
  float* bpart = (float*)(ws + WS_BPART);
  float* scal = (float*)(ws + WS_SCAL);
  float* hist = (float*)(ws + WS_HIST);

  kInit<<<1, 32, 0, stream>>>(y, scal);
  kA<<<NBLK_A, 256, 0, stream>>>(probs, y, unc, accb, bpart, scal);
  kR1<<<1, 256, 0, stream>>>(bpart, scal);
  kB<<<NBLK_B, 256, 0, stream>>>(unc, accb, scal, hist);
  kC<<<1, 128, 0, stream>>>(hist, scal, out);
}